// QLSTM_65481071403381
// MI455X (gfx1250) — compile-verified
//
#include <hip/hip_runtime.h>
#include <math.h>

typedef float v2f __attribute__((ext_vector_type(2)));
typedef float v8f __attribute__((ext_vector_type(8)));

#define TQ 128
#define BQ 256
#define DQ 64
#define HQ 10
#define KD 74  /* D + H */

// Single fused kernel: 16 blocks (one per 16-row batch tile), 128 threads
// (4 waves; wave g owns gate g's 16x16 WMMA tile). The full T=128 recurrence
// runs inside the kernel -- batch rows never couple across tiles.
//
// Quantum gate, collapsed analytically (product state + RZ phase-invariance +
// CNOT chain = prefix-XOR permutation):
//   qgate(pre, p)[b,w] = prod_{u<=w} cos(p[u,1]) * cos(p[u,0] + pre[b,u])
__global__ __launch_bounds__(128) void qlstm_fused(
    const float* __restrict__ inputs,
    const float* __restrict__ Wf, const float* __restrict__ Wi,
    const float* __restrict__ Wu, const float* __restrict__ Wo,
    const float* __restrict__ bf, const float* __restrict__ bi,
    const float* __restrict__ bu, const float* __restrict__ bo,
    const float* __restrict__ pf, const float* __restrict__ pi,
    const float* __restrict__ pu, const float* __restrict__ po,
    const float* __restrict__ lng, const float* __restrict__ lnb,
    float* __restrict__ ys, float* __restrict__ hout, float* __restrict__ cout) {
  __shared__ float wLds[4][16][80];   // zero-padded W per gate (W^T columns)
  __shared__ float combLds[16][80];   // [x_t | h_{t-1}] tile, zero-padded K
  __shared__ float fac[4][16][HQ];    // per-wire <Z> factors
  __shared__ float gact[4][16][HQ];   // activated gate values
  __shared__ float hLds[16][HQ];
  __shared__ float cLds[16][HQ];
  __shared__ float hraw[16][HQ];

  const int mtile = blockIdx.x;
  const int tid   = threadIdx.x;
  const int g     = tid >> 5;   // wave -> gate (f,i,u,o)
  const int lane  = tid & 31;

  const float* bbg = (g == 0) ? bf : (g == 1) ? bi : (g == 2) ? bu : bo;
  const float* pg  = (g == 0) ? pf : (g == 1) ? pi : (g == 2) ? pu : po;

  // ---- one-time init: zero pads, load weights, zero h/c ----
  for (int idx = tid; idx < 4 * 16 * 80; idx += 128) ((float*)wLds)[idx] = 0.f;
  for (int idx = tid; idx < 16 * 80; idx += 128) ((float*)combLds)[idx] = 0.f;
  for (int idx = tid; idx < 16 * HQ; idx += 128) {
    ((float*)hLds)[idx] = 0.f;
    ((float*)cLds)[idx] = 0.f;
  }
  __syncthreads();
  for (int idx = tid; idx < 4 * HQ * KD; idx += 128) {
    int gl = idx / (HQ * KD), rem = idx % (HQ * KD);
    const float* Wsrc = (gl == 0) ? Wf : (gl == 1) ? Wi : (gl == 2) ? Wu : Wo;
    wLds[gl][rem / KD][rem % KD] = Wsrc[rem];
  }
  __syncthreads();

  // per-lane invariants (full-precision cos, computed once)
  const int n    = lane & 15;  // output wire / B,D column
  const int m    = lane & 15;  // A-matrix row
  const int half = lane >> 4;  // K sub-pair selector
  const float bias = (n < HQ) ? bbg[n] : 0.f;
  const float px   = (n < HQ) ? pg[n * 3 + 0] : 0.f;
  const float cpy  = (n < HQ) ? cosf(pg[n * 3 + 1]) : 0.f;

  for (int t = 0; t < TQ; ++t) {
    // ---- stage comb = [x_t | h_prev] ----
    const float* xt = inputs + ((size_t)t * BQ + mtile * 16) * DQ;
#pragma unroll
    for (int rep = 0; rep < 2; ++rep) {
      int ii = rep * 128 + tid;
      int r = ii >> 4, c4 = ii & 15;
      *(float4*)&combLds[r][c4 * 4] = ((const float4*)(xt + r * DQ))[c4];
    }
    for (int idx = tid; idx < 16 * HQ; idx += 128) {
      int r = idx / HQ, c = idx % HQ;
      combLds[r][DQ + c] = hLds[r][c];
    }
    if (t + 1 < TQ) {  // warm next step's x tile (global_prefetch_b8)
      const char* nx = (const char*)(inputs + ((size_t)(t + 1) * BQ + mtile * 16) * DQ);
      __builtin_prefetch(nx + tid * 32, 0, 3);
    }
    __syncthreads();

    // ---- GEMM: wave g -> 16(batch) x 16(wire) tile, K=76 in 19 wmma ----
    v8f acc = {0.f, 0.f, 0.f, 0.f, 0.f, 0.f, 0.f, 0.f};
#pragma unroll
    for (int kk = 0; kk < 19; ++kk) {
      int kb = kk * 4 + half * 2;
      v2f a = *(const v2f*)&combLds[m][kb];
      v2f b = *(const v2f*)&wLds[g][n][kb];
      acc = __builtin_amdgcn_wmma_f32_16x16x4_f32(false, a, false, b,
                                                  (short)0, acc, false, false);
    }
    // per-wire factor: cos(py) * cos(px + pre); args are O(1) -> v_cos_f32
    if (n < HQ) {
#pragma unroll
      for (int v = 0; v < 8; ++v) {
        int row = v + 8 * half;  // C/D layout: VGPR v <-> M = v / v+8
        fac[g][row][n] = cpy * __cosf(px + acc[v] + bias);
      }
    }
    __syncthreads();

    // ---- prefix product over wires + activation: (gate,row) per thread ----
    if (tid < 64) {
      int g2 = tid >> 4, row = tid & 15;
      float z = 1.f;
      for (int w = 0; w < HQ; ++w) {
        z *= fac[g2][row][w];
        gact[g2][row][w] = (g2 == 2) ? tanhf(z) : (1.f / (1.f + __expf(-z)));
      }
    }
    __syncthreads();

    // ---- cell update + residual ----
    for (int idx = tid; idx < 16 * HQ; idx += 128) {
      int row = idx / HQ, w = idx % HQ;
      float fv = gact[0][row][w], iv = gact[1][row][w];
      float gv = gact[2][row][w], ov = gact[3][row][w];
      float cn = fv * cLds[row][w] + iv * gv;
      cLds[row][w] = cn;
      hraw[row][w] = ov * tanhf(cn) + hLds[row][w];
    }
    __syncthreads();

    // ---- LayerNorm over H=10, write ys and new h ----
    for (int idx = tid; idx < 16 * HQ; idx += 128) {
      int row = idx / HQ, w = idx % HQ;
      float mu = 0.f;
      for (int u = 0; u < HQ; ++u) mu += hraw[row][u];
      mu *= 0.1f;
      float var = 0.f;
      for (int u = 0; u < HQ; ++u) { float d = hraw[row][u] - mu; var += d * d; }
      var *= 0.1f;
      float hv = (hraw[row][w] - mu) * rsqrtf(var + 1e-5f) * lng[w] + lnb[w];
      hLds[row][w] = hv;
      ys[((size_t)t * BQ + mtile * 16 + row) * HQ + w] = hv;
    }
    __syncthreads();
  }

  // ---- final carries (hx, cx) ----
  for (int idx = tid; idx < 16 * HQ; idx += 128) {
    int row = idx / HQ, w = idx % HQ;
    hout[(mtile * 16 + row) * HQ + w] = hLds[row][w];
    cout[(mtile * 16 + row) * HQ + w] = cLds[row][w];
  }
}

extern "C" void kernel_launch(void* const* d_in, const int* in_sizes, int n_in,
                              void* d_out, int out_size, void* d_ws, size_t ws_size,
                              hipStream_t stream) {
  const float* inputs = (const float*)d_in[0];
  const float* Wf = (const float*)d_in[1];
  const float* bf = (const float*)d_in[2];
  const float* pf = (const float*)d_in[3];
  const float* Wi = (const float*)d_in[4];
  const float* bi = (const float*)d_in[5];
  const float* pi = (const float*)d_in[6];
  const float* Wu = (const float*)d_in[7];
  const float* bu = (const float*)d_in[8];
  const float* pu = (const float*)d_in[9];
  const float* Wo = (const float*)d_in[10];
  const float* bo = (const float*)d_in[11];
  const float* po = (const float*)d_in[12];
  const float* lng = (const float*)d_in[13];
  const float* lnb = (const float*)d_in[14];

  float* out  = (float*)d_out;
  float* ys   = out;                        // (T,B,H)
  float* hout = out + (size_t)TQ * BQ * HQ; // hx
  float* cout = hout + BQ * HQ;             // cx

  qlstm_fused<<<BQ / 16, 128, 0, stream>>>(
      inputs, Wf, Wi, Wu, Wo, bf, bi, bu, bo,
      pf, pi, pu, po, lng, lnb, ys, hout, cout);
}